// _GenericPIC2_53377853554802
// MI455X (gfx1250) — compile-verified
//
#include <hip/hip_runtime.h>
#include <hip/hip_bf16.h>
#include <math.h>

// Problem dimensions (fixed by the reference)
#define B_  512
#define D_  8
#define W_  128
#define N_  64
#define K_  64

#define T_COUP 0.95f
#define A_LOSS 0.99f
#define N_EFF  2.4f
#define R0_    5.0f
#define TWO_PI 6.28318530717958647692f

typedef __attribute__((ext_vector_type(2))) float v2f;
typedef __attribute__((ext_vector_type(8))) float v8f;

static __device__ __forceinline__ v8f wmma4(v2f a, v2f b, v8f c) {
    // D = A(16x4) * B(4x16) + C(16x16), fp32 in / fp32 out
    return __builtin_amdgcn_wmma_f32_16x16x4_f32(
        /*neg_a=*/false, a, /*neg_b=*/false, b,
        /*c_mod=*/(short)0, c, /*reuse_a=*/false, /*reuse_b=*/false);
}

// ---------------------------------------------------------------------------
// Kernel 1: build M[d,w] = prod_k ( C * diag(t_k) )  via  M <- (M @ C) * diag(t_k)
// Complex matmul with Gauss 3-multiply: P=Ar*Br, Q=Ai*Bi, S=(Ar+Ai)(Br+Bi)
//   Re = P - Q ; Im = S - P - Q
// One workgroup (256 threads = 8 waves) per (d,w) pair. 1024 workgroups.
// Each wave owns 2 complex 16x16 output tiles sharing the same column tile,
// so B fragments (C matrix) are loaded once per k-step for both tiles.
// ---------------------------------------------------------------------------
__global__ __launch_bounds__(256) void pic_build_M(
    const float* __restrict__ vals,   // [K, D, N]
    const float* __restrict__ wls,    // [W]
    float* __restrict__ Mr_g,         // [D*W, 64, 64]
    float* __restrict__ Mi_g)         // [D*W, 64, 64]
{
    // C in row-pair layout: CrP[p][2j] = C[2p][j], CrP[p][2j+1] = C[2p+1][j]
    __shared__ float CrP[32][130];
    __shared__ float CiP[32][130];
    __shared__ float CsP[32][130];     // Cr + Ci (for Gauss term)
    __shared__ float Ms[2][2][64][66]; // [buf][re/im][row][col], padded
    __shared__ float tRe[64], tIm[64];

    const int tid  = threadIdx.x;
    const int dw   = blockIdx.x;
    const int d    = dw >> 7;          // dw / W_
    const int w    = dw & (W_ - 1);
    const float wl = wls[w];

    // ---- build DFT matrix C = exp(-2*pi*i*l*j/N)/sqrt(N) in LDS ----
    for (int idx = tid; idx < 32 * 64; idx += 256) {
        int p = idx >> 6, j = idx & 63;
        int l0 = 2 * p, l1 = 2 * p + 1;
        int m0 = (l0 * j) & 63;        // exact periodic reduction
        int m1 = (l1 * j) & 63;
        float a0 = -TWO_PI * (float)m0 * (1.0f / 64.0f);
        float a1 = -TWO_PI * (float)m1 * (1.0f / 64.0f);
        float c0, s0, c1, s1;
        sincosf(a0, &s0, &c0);
        sincosf(a1, &s1, &c1);
        c0 *= 0.125f; s0 *= 0.125f; c1 *= 0.125f; s1 *= 0.125f;
        CrP[p][2 * j] = c0;       CrP[p][2 * j + 1] = c1;
        CiP[p][2 * j] = s0;       CiP[p][2 * j + 1] = s1;
        CsP[p][2 * j] = c0 + s0;  CsP[p][2 * j + 1] = c1 + s1;
    }
    // ---- M = identity in buffer 0 ----
    for (int idx = tid; idx < 64 * 64; idx += 256) {
        int i = idx >> 6, l = idx & 63;
        Ms[0][0][i][l] = (i == l) ? 1.0f : 0.0f;
        Ms[0][1][i][l] = 0.0f;
    }
    __syncthreads();

    const int lane = tid & 31;
    const int wave = tid >> 5;
    const int half = lane >> 4;
    const int l16  = lane & 15;
    const int tj   = wave & 3;         // column tile (same for both tiles)
    const int ti0  = wave >> 2;        // row tiles: ti0 and ti0+2
    const int row0 = ti0 * 16 + l16;
    const int row1 = row0 + 32;
    const int jcol = tj * 16 + l16;

    for (int k = 0; k < K_; ++k) {
        const int cur = k & 1;
        const int nxt = cur ^ 1;

        // ---- per-step ring response t[n] (64 lanes) ----
        if (tid < 64) {
            int n = tid;
            float v   = vals[(k * D_ + d) * N_ + n];
            float Ln  = TWO_PI * R0_ * (1.0f + 0.01f * (float)n);
            float phi = (TWO_PI * N_EFF) * (Ln / wl) + TWO_PI * v;
            float sp, cp;
            sincosf(phi, &sp, &cp);
            float nr = T_COUP - A_LOSS * cp;
            float ni = -A_LOSS * sp;
            float dr = 1.0f - T_COUP * A_LOSS * cp;
            float di = -T_COUP * A_LOSS * sp;
            float inv = 1.0f / (dr * dr + di * di);
            tRe[n] = (nr * dr + ni * di) * inv;
            tIm[n] = (ni * dr - nr * di) * inv;
        }
        __syncthreads();

        // ---- both tiles in one fused k-loop: 6 WMMA / 7 LDS loads per step ----
        v8f P0 = {}, Q0 = {}, S0 = {};
        v8f P1 = {}, Q1 = {}, S1 = {};
#pragma unroll 4
        for (int kb = 0; kb < 16; ++kb) {
            const int col  = kb * 4 + 2 * half;
            const int prow = kb * 2 + half;
            v2f br  = *(const v2f*)&CrP[prow][2 * jcol];
            v2f bi  = *(const v2f*)&CiP[prow][2 * jcol];
            v2f bs  = *(const v2f*)&CsP[prow][2 * jcol];
            v2f ar0 = *(const v2f*)&Ms[cur][0][row0][col];
            v2f ai0 = *(const v2f*)&Ms[cur][1][row0][col];
            v2f ar1 = *(const v2f*)&Ms[cur][0][row1][col];
            v2f ai1 = *(const v2f*)&Ms[cur][1][row1][col];
            v2f as0 = ar0 + ai0;
            v2f as1 = ar1 + ai1;
            P0 = wmma4(ar0, br, P0);
            Q0 = wmma4(ai0, bi, Q0);
            S0 = wmma4(as0, bs, S0);
            P1 = wmma4(ar1, br, P1);
            Q1 = wmma4(ai1, bi, Q1);
            S1 = wmma4(as1, bs, S1);
        }

        // ---- combine Gauss terms, scale columns by t, write M_nxt ----
        const float tr  = tRe[jcol];
        const float tii = tIm[jcol];
#pragma unroll
        for (int r = 0; r < 8; ++r) {
            const int orow0 = ti0 * 16 + r + 8 * half;
            float xr0 = P0[r] - Q0[r];
            float xi0 = S0[r] - P0[r] - Q0[r];
            Ms[nxt][0][orow0][jcol] = xr0 * tr - xi0 * tii;
            Ms[nxt][1][orow0][jcol] = xr0 * tii + xi0 * tr;
            const int orow1 = orow0 + 32;
            float xr1 = P1[r] - Q1[r];
            float xi1 = S1[r] - P1[r] - Q1[r];
            Ms[nxt][0][orow1][jcol] = xr1 * tr - xi1 * tii;
            Ms[nxt][1][orow1][jcol] = xr1 * tii + xi1 * tr;
        }
        __syncthreads();
    }

    // final result sits in buffer (K_ % 2 == 0 ? 0 : 1) == 0
    for (int idx = tid; idx < 64 * 64; idx += 256) {
        int i = idx >> 6, l = idx & 63;
        Mr_g[(size_t)dw * 4096 + idx] = Ms[0][0][i][l];
        Mi_g[(size_t)dw * 4096 + idx] = Ms[0][1][i][l];
    }
}

// ---------------------------------------------------------------------------
// Kernel 2: out[b,d,w,i] = sum_j M[d,w,i,j] * x[b,d,w,j]
// One workgroup per (d,w, 64-wide batch tile). 1024*8 = 8192 workgroups.
// ---------------------------------------------------------------------------
__global__ __launch_bounds__(256) void pic_apply(
    const float* __restrict__ xr_g,   // [B, D, W, N]
    const float* __restrict__ xi_g,
    const float* __restrict__ Mr_g,   // [D*W, 64, 64]
    const float* __restrict__ Mi_g,
    float* __restrict__ out)          // [B, D, W, N] complex64 interleaved
{
    __shared__ float Msr[64][66], Msi[64][66];              // M tile
    __shared__ float Xr[64][66], Xi[64][66], Xs[64][66];    // x tile [b][j]

    const int tid = threadIdx.x;
    const int bid = blockIdx.x;
    const int dw  = bid >> 3;
    const int bt  = bid & 7;
    const int d   = dw >> 7;
    const int w   = dw & (W_ - 1);
    const int b0  = bt * 64;

    // ---- stage M and x (coalesced) ----
    for (int idx = tid; idx < 64 * 64; idx += 256) {
        int i = idx >> 6, l = idx & 63;
        Msr[i][l] = Mr_g[(size_t)dw * 4096 + idx];
        Msi[i][l] = Mi_g[(size_t)dw * 4096 + idx];
    }
    for (int idx = tid; idx < 64 * 64; idx += 256) {
        int b = idx >> 6, j = idx & 63;
        int g = (((b0 + b) * D_ + d) * W_ + w) * N_ + j;
        float vr = xr_g[g], vi = xi_g[g];
        Xr[b][j] = vr; Xi[b][j] = vi; Xs[b][j] = vr + vi;
    }
    __syncthreads();

    const int lane = tid & 31;
    const int wave = tid >> 5;
    const int half = lane >> 4;
    const int l16  = lane & 15;
    const int tj   = wave & 3;            // batch tile (same for both tiles)
    const int ti0  = wave >> 2;           // output-row tiles: ti0 and ti0+2
    const int row0 = ti0 * 16 + l16;      // i for A fragments
    const int row1 = row0 + 32;
    const int bcol = tj * 16 + l16;       // b for B fragments

    v8f P0 = {}, Q0 = {}, S0 = {};
    v8f P1 = {}, Q1 = {}, S1 = {};
#pragma unroll 4
    for (int kb = 0; kb < 16; ++kb) {
        const int col = kb * 4 + 2 * half;    // contracted j
        v2f br  = *(const v2f*)&Xr[bcol][col];
        v2f bi  = *(const v2f*)&Xi[bcol][col];
        v2f bs  = *(const v2f*)&Xs[bcol][col];
        v2f ar0 = *(const v2f*)&Msr[row0][col];
        v2f ai0 = *(const v2f*)&Msi[row0][col];
        v2f ar1 = *(const v2f*)&Msr[row1][col];
        v2f ai1 = *(const v2f*)&Msi[row1][col];
        v2f as0 = ar0 + ai0;
        v2f as1 = ar1 + ai1;
        P0 = wmma4(ar0, br, P0);
        Q0 = wmma4(ai0, bi, Q0);
        S0 = wmma4(as0, bs, S0);
        P1 = wmma4(ar1, br, P1);
        Q1 = wmma4(ai1, bi, Q1);
        S1 = wmma4(as1, bs, S1);
    }
    __syncthreads();   // all waves done reading X tiles

    // ---- combine + transpose through LDS (reuse Xr/Xi as Out[b][i]) ----
#pragma unroll
    for (int r = 0; r < 8; ++r) {
        const int b = tj * 16 + l16;
        const int i0 = ti0 * 16 + r + 8 * half;
        Xr[b][i0] = P0[r] - Q0[r];
        Xi[b][i0] = S0[r] - P0[r] - Q0[r];
        const int i1 = i0 + 32;
        Xr[b][i1] = P1[r] - Q1[r];
        Xi[b][i1] = S1[r] - P1[r] - Q1[r];
    }
    __syncthreads();

    // ---- coalesced interleaved complex store ----
    for (int idx = tid; idx < 64 * 64; idx += 256) {
        int b = idx >> 6, i = idx & 63;
        size_t g = ((size_t)(((b0 + b) * D_ + d) * W_ + w) * N_ + i) * 2;
        v2f o; o.x = Xr[b][i]; o.y = Xi[b][i];
        *(v2f*)&out[g] = o;
    }
}

extern "C" void kernel_launch(void* const* d_in, const int* in_sizes, int n_in,
                              void* d_out, int out_size, void* d_ws, size_t ws_size,
                              hipStream_t stream) {
    const float* x_real = (const float*)d_in[0];
    const float* x_imag = (const float*)d_in[1];
    const float* vals   = (const float*)d_in[2];
    const float* wls    = (const float*)d_in[3];
    float* out = (float*)d_out;

    // Workspace: M real plane then imag plane, each D*W*64*64 floats (16 MB)
    float* Mr = (float*)d_ws;
    float* Mi = Mr + (size_t)D_ * W_ * N_ * N_;

    pic_build_M<<<D_ * W_, 256, 0, stream>>>(vals, wls, Mr, Mi);
    pic_apply<<<D_ * W_ * (B_ / 64), 256, 0, stream>>>(x_real, x_imag, Mr, Mi, out);
}